// HistoryEncoder_12584254177378
// MI455X (gfx1250) — compile-verified
//
#include <hip/hip_runtime.h>
#include <hip/hip_bf16.h>
#include <stdint.h>

// Problem constants (match reference)
#define N_EV     400000
#define NLOCS    100000
#define NTIMES   168
#define D_LOC    64
#define D_TIME   32
#define D_USER   64
#define D_EMB    160
#define D_OUT    256
#define KEYS     16800000      // NLOCS * NTIMES
#define SCAN_B   4096          // elements per scan block (256 thr x 16)
#define NB0      4102          // ceil(KEYS / SCAN_B)
#define KEYS_PAD (NB0 * SCAN_B)

typedef __attribute__((ext_vector_type(2))) float v2f;
typedef __attribute__((ext_vector_type(8))) float v8f;
typedef unsigned int u32;

// ---------------------------------------------------------------- zero
__global__ void k_zero(u32* __restrict__ p, size_t n) {
  size_t i = (size_t)blockIdx.x * blockDim.x + threadIdx.x;
  size_t stride = (size_t)gridDim.x * blockDim.x;
  for (; i < n; i += stride) p[i] = 0u;
}

// ------------------------------------------------- key + histogram
__global__ void k_hist(const int* __restrict__ x, const int* __restrict__ t,
                       u32* __restrict__ keys, u32* __restrict__ cnt) {
  int i = blockIdx.x * blockDim.x + threadIdx.x;
  if (i < N_EV) {
    u32 key = (u32)x[i] * (u32)NTIMES + (u32)t[i];
    keys[i] = key;
    atomicAdd(&cnt[key], 1u);
  }
}

// -------------------------------- scan L0: per-block exclusive scan of flags
__global__ void k_scan_l0(const u32* __restrict__ cnt, u32* __restrict__ ranks,
                          u32* __restrict__ bsums) {
  __shared__ u32 s[256];
  const int t = threadIdx.x;
  const size_t base = (size_t)blockIdx.x * SCAN_B + (size_t)t * 16;
  u32 f[16];
  u32 tot = 0;
  const uint4* in4 = (const uint4*)(cnt + base);
#pragma unroll
  for (int j = 0; j < 4; ++j) {
    uint4 v = in4[j];
    f[4*j+0] = v.x ? 1u : 0u; f[4*j+1] = v.y ? 1u : 0u;
    f[4*j+2] = v.z ? 1u : 0u; f[4*j+3] = v.w ? 1u : 0u;
    tot += f[4*j+0] + f[4*j+1] + f[4*j+2] + f[4*j+3];
  }
  s[t] = tot;
  __syncthreads();
#pragma unroll
  for (int off = 1; off < 256; off <<= 1) {
    u32 v = (t >= off) ? s[t - off] : 0u;
    __syncthreads();
    s[t] += v;
    __syncthreads();
  }
  u32 run = s[t] - tot;               // exclusive base for this thread
  uint4* out4 = (uint4*)(ranks + base);
#pragma unroll
  for (int j = 0; j < 4; ++j) {
    uint4 o;
    o.x = run; run += f[4*j+0];
    o.y = run; run += f[4*j+1];
    o.z = run; run += f[4*j+2];
    o.w = run; run += f[4*j+3];
    out4[j] = o;
  }
  if (t == 255) bsums[blockIdx.x] = s[255];
}

// -------------------------------- scan L1: exclusive scan of block sums
__global__ void k_scan_l1(u32* __restrict__ bsums, u32* __restrict__ mOut) {
  __shared__ u32 s[1024];
  const int t = threadIdx.x;
  u32 loc[5];
  u32 tot = 0;
#pragma unroll
  for (int j = 0; j < 5; ++j) {
    int i = t * 5 + j;
    u32 v = (i < NB0) ? bsums[i] : 0u;
    loc[j] = v; tot += v;
  }
  s[t] = tot;
  __syncthreads();
  for (int off = 1; off < 1024; off <<= 1) {
    u32 v = (t >= off) ? s[t - off] : 0u;
    __syncthreads();
    s[t] += v;
    __syncthreads();
  }
  u32 run = s[t] - tot;
#pragma unroll
  for (int j = 0; j < 5; ++j) {
    int i = t * 5 + j;
    if (i < NB0) bsums[i] = run;
    run += loc[j];
  }
  if (t == 1023) *mOut = s[1023];     // M = number of unique keys
}

// ---------------------------- emit sorted unique keys + counts
__global__ void k_emit(const u32* __restrict__ cnt, const u32* __restrict__ ranks,
                       const u32* __restrict__ bsums,
                       u32* __restrict__ ukey, u32* __restrict__ ucnt) {
  size_t idx = (size_t)blockIdx.x * blockDim.x + threadIdx.x;
  if (idx < (size_t)KEYS) {
    u32 c = cnt[idx];
    if (c) {
      u32 r = ranks[idx] + bsums[idx >> 12];
      ukey[r] = (u32)idx;
      ucnt[r] = c;
    }
  }
}

// ---------------------------- scatter-add user embeddings per unique key
__global__ void k_scatter(const u32* __restrict__ keys, const u32* __restrict__ ranks,
                          const u32* __restrict__ bsums, const int* __restrict__ uu,
                          const float* __restrict__ user_t, float* __restrict__ usum) {
  size_t tid = (size_t)blockIdx.x * blockDim.x + threadIdx.x;
  if (tid < (size_t)N_EV * D_USER) {
    size_t i = tid >> 6;
    int c = (int)(tid & 63);
    u32 key = keys[i];
    u32 r = ranks[key] + bsums[key >> 12];
    atomicAdd(&usum[(size_t)r * D_USER + c],
              user_t[(size_t)uu[i] * D_USER + c]);
  }
}

// ---------------------------- WMMA GEMM: out[r,:] = (loc||time||umean) @ W + b
// One block = 16 output rows (one M-tile), 8 waves; each wave does 2 col-tiles.
// K = 160 accumulated with V_WMMA_F32_16X16X4_F32 (exact f32).
__global__ void __launch_bounds__(256)
k_gemm(const u32* __restrict__ ukey, const u32* __restrict__ ucnt,
       const float* __restrict__ usum, const u32* __restrict__ mPtr,
       const float* __restrict__ loc_t, const float* __restrict__ time_t,
       const float* __restrict__ W, const float* __restrict__ bias,
       float* __restrict__ out) {
  __shared__ float A[16][168];                 // padded stride vs 64 banks
  const u32 M = *mPtr;
  const int r0 = blockIdx.x * 16;

  // Build the 16 x 160 A tile cooperatively (concat + grouped mean).
  for (int e = threadIdx.x; e < 16 * D_EMB; e += 256) {
    int row = e / D_EMB, c = e % D_EMB;
    u32 r = (u32)(r0 + row);
    float v = 0.f;
    if (r < M) {
      u32 key = ukey[r];
      if (c < D_LOC) {
        v = loc_t[(size_t)(key / NTIMES) * D_LOC + c];
      } else if (c < D_LOC + D_TIME) {
        v = time_t[(size_t)(key % NTIMES) * D_TIME + (c - D_LOC)];
      } else {
        u32 cc = ucnt[r]; if (cc < 1u) cc = 1u;
        v = usum[(size_t)r * D_USER + (c - D_LOC - D_TIME)] / (float)cc;
      }
    }
    A[row][c] = v;
  }
  __syncthreads();

  const int lane = threadIdx.x & 31;
  const int wid  = threadIdx.x >> 5;
  const int hs   = lane >> 4;                  // half-wave select (K offset 2)
  const int lr   = lane & 15;                  // row (A) / col (B,C) index

#pragma unroll
  for (int ct = 0; ct < 2; ++ct) {
    const int col = (wid * 2 + ct) * 16 + lr;  // output column for this lane
    const float bv = bias[col];
    v8f acc = {0.f, 0.f, 0.f, 0.f, 0.f, 0.f, 0.f, 0.f};
#pragma unroll
    for (int k0 = 0; k0 < D_EMB; k0 += 4) {
      v2f a, b;
      // A 16x4 frag: VGPR j holds K = k0 + j + 2*hs for row lr
      a.x = A[lr][k0 + 2 * hs + 0];
      a.y = A[lr][k0 + 2 * hs + 1];
      // B 4x16 frag: VGPR j holds K-row k0 + j + 2*hs, N = col
      b.x = W[(size_t)(k0 + 2 * hs + 0) * D_OUT + col];
      b.y = W[(size_t)(k0 + 2 * hs + 1) * D_OUT + col];
      acc = __builtin_amdgcn_wmma_f32_16x16x4_f32(
          false, a, false, b, (short)0, acc, false, false);
    }
#pragma unroll
    for (int i = 0; i < 8; ++i) {
      int row = r0 + i + 8 * hs;               // C/D layout: VGPR i -> M=i(+8)
      float v = ((u32)row < M) ? (acc[i] + bv) : 0.f;
      out[(size_t)row * D_OUT + col] = v;
    }
  }
}

// ---------------------------------------------------------------- launch
extern "C" void kernel_launch(void* const* d_in, const int* in_sizes, int n_in,
                              void* d_out, int out_size, void* d_ws, size_t ws_size,
                              hipStream_t stream) {
  const int*   x      = (const int*)d_in[0];
  const int*   t      = (const int*)d_in[1];
  const int*   u      = (const int*)d_in[2];
  const float* loc_t  = (const float*)d_in[3];
  const float* time_t = (const float*)d_in[4];
  const float* user_t = (const float*)d_in[5];
  const float* W      = (const float*)d_in[6];
  const float* bias   = (const float*)d_in[7];
  float* out = (float*)d_out;
  (void)in_sizes; (void)n_in; (void)out_size; (void)ws_size;

  // Workspace carve-out (~242 MB total), 256B-aligned slabs.
  char* ws = (char*)d_ws;
  size_t off = 0;
  auto alloc = [&](size_t bytes) -> void* {
    void* p = ws + off;
    off += (bytes + 255) & ~(size_t)255;
    return p;
  };
  u32*   cnt   = (u32*)alloc((size_t)KEYS_PAD * 4);
  u32*   ranks = (u32*)alloc((size_t)KEYS_PAD * 4);
  u32*   bsums = (u32*)alloc((size_t)NB0 * 4);
  u32*   keys  = (u32*)alloc((size_t)N_EV * 4);
  u32*   ukey  = (u32*)alloc((size_t)N_EV * 4);
  u32*   ucnt  = (u32*)alloc((size_t)N_EV * 4);
  u32*   mPtr  = (u32*)alloc(4);
  float* usum  = (float*)alloc((size_t)N_EV * D_USER * 4);

  k_zero<<<2048, 256, 0, stream>>>(cnt, (size_t)KEYS_PAD);
  k_zero<<<2048, 256, 0, stream>>>((u32*)usum, (size_t)N_EV * D_USER);

  k_hist<<<(N_EV + 255) / 256, 256, 0, stream>>>(x, t, keys, cnt);

  k_scan_l0<<<NB0, 256, 0, stream>>>(cnt, ranks, bsums);
  k_scan_l1<<<1, 1024, 0, stream>>>(bsums, mPtr);

  k_emit<<<(KEYS + 255) / 256, 256, 0, stream>>>(cnt, ranks, bsums, ukey, ucnt);

  k_scatter<<<(int)(((size_t)N_EV * D_USER + 255) / 256), 256, 0, stream>>>(
      keys, ranks, bsums, u, user_t, usum);

  k_gemm<<<N_EV / 16, 256, 0, stream>>>(ukey, ucnt, usum, mPtr,
                                        loc_t, time_t, W, bias, out);
}